// VN_DGCNN_44478681317785
// MI455X (gfx1250) — compile-verified
//
#include <hip/hip_runtime.h>
#include <math.h>

// ---------------------------------------------------------------------------
// VN-DGCNN forward for MI455X (gfx1250, wave32).
// fp32 end-to-end; matrix work on V_WMMA_F32_16X16X4_F32.
// KNN distance strips are kept in LDS and top-k'd in place (never spilled).
// [W;Wd] are packed into one A matrix so each big activation tensor is read
// once per vn_block. K dimensions are compile-time (templates) so fragment
// indexing is static: no v_movrels, fully unrolled load->wmma streams.
// ---------------------------------------------------------------------------

#define EPSV   1e-6f
#define BNEPS  1e-5f
#define NEGS   0.2f

typedef float v2f __attribute__((ext_vector_type(2)));
typedef float v8f __attribute__((ext_vector_type(8)));

__device__ __forceinline__ v8f wmma_f32x4(v2f a, v2f b, v8f c) {
  // D = A(16x4,f32) * B(4x16,f32) + C(16x16,f32)
  return __builtin_amdgcn_wmma_f32_16x16x4_f32(false, a, false, b, (short)0, c,
                                               false, false);
}

// ---------------------------------------------------------------- row |x|^2
__global__ void row_sumsq_kernel(const float* __restrict__ x, float* __restrict__ xx,
                                 int D, int N, long bStride, int Btot) {
  long t = (long)blockIdx.x * blockDim.x + threadIdx.x;
  long total = (long)Btot * N;
  if (t >= total) return;
  int b = (int)(t / N), n = (int)(t % N);
  const float* p = x + (long)b * bStride + n;
  float s = 0.f;
  for (int d = 0; d < D; ++d) { float v = p[(long)d * N]; s += v * v; }
  xx[t] = s;
}

// ------------------------------------------------- fused WMMA dist + top-20
// Block = 128 threads (4 waves). Each block owns 16 rows of one batch:
// phase 1: WMMA-accumulate neg_dist strip (16 x N) into LDS,
// phase 2: 16 lanes do in-register top-20 (statically indexed bubble).
// DCH is compile-time -> afr[] statically indexed, K-loop fully unrolled.
template <int DCH>
__global__ void knn_topk_kernel(const float* __restrict__ x, const float* __restrict__ xx,
                                int* __restrict__ knn, int N, long bStride) {
  extern __shared__ float lds[];               // 16 * N floats
  constexpr int KSTEPS = (DCH + 3) / 4;
  const int b = blockIdx.y;
  const int r0 = blockIdx.x * 16;
  const int lane = threadIdx.x & 31;
  const int wave = threadIdx.x >> 5;
  const int nWaves = blockDim.x >> 5;
  const int half = lane >> 4;                  // 0: lanes 0-15, 1: lanes 16-31
  const int l16 = lane & 15;
  const float* xb = x + (long)b * bStride;

  // Preload A fragments for the 16 rows of this block.
  v2f afr[KSTEPS];
#pragma unroll
  for (int ks = 0; ks < KSTEPS; ++ks) {
    v2f a{}; a.x = 0.f; a.y = 0.f;
    const int k0 = ks * 4 + half * 2;
    const int row = r0 + l16;
    if (k0 < DCH)     a.x = xb[(long)k0 * N + row];
    if (k0 + 1 < DCH) a.y = xb[(long)(k0 + 1) * N + row];
    afr[ks] = a;
  }

  const int nTiles = N >> 4;
  for (int t = wave; t < nTiles; t += nWaves) {
    const int c0 = t * 16;
    v8f acc = {};
#pragma unroll
    for (int ks = 0; ks < KSTEPS; ++ks) {
      const int kb = ks * 4;
      const int kx = kb + half, ky = kb + 2 + half;
      v2f bf{}; bf.x = 0.f; bf.y = 0.f;
      if (kx < DCH) bf.x = xb[(long)kx * N + c0 + l16];
      if (ky < DCH) bf.y = xb[(long)ky * N + c0 + l16];
      acc = wmma_f32x4(afr[ks], bf, acc);
    }
    const float xxc = xx[(long)b * N + c0 + l16];
#pragma unroll
    for (int v = 0; v < 8; ++v) {
      int row = v + half * 8;
      float nd = 2.f * acc[v] - xx[(long)b * N + r0 + row] - xxc;
      lds[row * N + c0 + l16] = nd;
    }
  }
  __syncthreads();

  if (threadIdx.x < 16) {
    const int row = threadIdx.x;
    float bv[20]; int bi[20];
#pragma unroll
    for (int j = 0; j < 20; ++j) { bv[j] = -3.0e38f; bi[j] = 0; }
    const float* rp = lds + row * N;
    for (int n = 0; n < N; ++n) {
      float v = rp[n];
      if (v > bv[19]) {
        bv[19] = v; bi[19] = n;
#pragma unroll
        for (int j = 18; j >= 0; --j) {
          if (bv[j] < bv[j + 1]) {
            float tv = bv[j]; bv[j] = bv[j + 1]; bv[j + 1] = tv;
            int ti = bi[j]; bi[j] = bi[j + 1]; bi[j + 1] = ti;
          }
        }
      }
    }
    int* op = knn + ((long)b * N + r0 + row) * 20;
#pragma unroll
    for (int j = 0; j < 20; ++j) op[j] = bi[j];
  }
}

// -------------------------------------------------- graph feature (gather)
// out (B, 2*Ci, 3, N, Kn): first Ci channels = x[j]-x[n], last Ci = x[n].
__global__ void graph_feature_kernel(const float* __restrict__ x, const int* __restrict__ knn,
                                     float* __restrict__ H0, int Ci, int N, int Kn,
                                     long xBS, long total) {
  long t = (long)blockIdx.x * blockDim.x + threadIdx.x;
  if (t >= total) return;
  int kk = (int)(t % Kn); long r = t / Kn;
  int n  = (int)(r % N);  r /= N;
  int c  = (int)(r % 3);  r /= 3;
  int i2 = (int)(r % (2 * Ci));
  int b  = (int)(r / (2 * Ci));
  float val;
  if (i2 < Ci) {
    int j = knn[((long)b * N + n) * Kn + kk];
    const float* xr = x + (long)b * xBS + (long)(i2 * 3 + c) * N;
    val = xr[j] - xr[n];
  } else {
    val = x[(long)b * xBS + (long)((i2 - Ci) * 3 + c) * N + n];
  }
  H0[t] = val;
}

// -------------------------------------------------------------- pack [W;Wd]
__global__ void packA_kernel(const float* __restrict__ W, const float* __restrict__ Wd,
                             float* __restrict__ A, int Co, int Cd, int K) {
  int t = blockIdx.x * blockDim.x + threadIdx.x;
  int total = (Co + Cd) * K;
  if (t >= total) return;
  int r = t / K, k = t % K;
  A[t] = (r < Co) ? W[r * K + k] : Wd[(r - Co) * K + k];
}

// ---------------------------------------------------- batched fp32 WMMA GEMM
// C[b] (M x Ncols, row major, ldc=Ncols) = A (M x K, shared) * B[b] (K x Ncols)
// KD is compile-time: tail guards fold, small-K loops fully unroll.
template <int KD>
__global__ void gemm_wmma_kernel(const float* __restrict__ A, const float* __restrict__ Bm,
                                 float* __restrict__ Cm, int M, long Ncols,
                                 long ldb, long bBS, long cBS) {
  constexpr int KSTEPS = (KD + 3) / 4;
  const int b = blockIdx.z;
  const int m0 = blockIdx.y * 16;
  const int lane = threadIdx.x & 31;
  const int wave = threadIdx.x >> 5;
  const int half = lane >> 4;
  const int l16 = lane & 15;
  const long c0 = (long)blockIdx.x * 128 + wave * 16;
  if (c0 >= Ncols) return;
  const float* Bb = Bm + (long)b * bBS;
  float* Cb = Cm + (long)b * cBS;
  const long col = c0 + l16;
  const bool cin = col < Ncols;

  const int arow = m0 + l16;
  const bool ain = arow < M;
  const float* Ar = A + (long)arow * KD;

  v8f acc = {};
  for (int ks = 0; ks < KSTEPS; ++ks) {
    const int kb = ks * 4;
    v2f a{}; a.x = 0.f; a.y = 0.f;
    {
      const int k0 = kb + half * 2;
      if (ain && k0 < KD)     a.x = Ar[k0];
      if (ain && k0 + 1 < KD) a.y = Ar[k0 + 1];
    }
    v2f bf{}; bf.x = 0.f; bf.y = 0.f;
    {
      const int kx = kb + half, ky = kb + 2 + half;
      if (cin && kx < KD) bf.x = Bb[(long)kx * ldb + col];
      if (cin && ky < KD) bf.y = Bb[(long)ky * ldb + col];
    }
    if (ks + 1 < KSTEPS)  // pull next K-rows of B through the cache path
      __builtin_prefetch(&Bb[(long)(kb + 4) * ldb + c0], 0, 0);
    acc = wmma_f32x4(a, bf, acc);
  }
  if (cin) {
#pragma unroll
    for (int v = 0; v < 8; ++v) {
      int row = m0 + v + half * 8;
      if (row < M) Cb[(long)row * Ncols + col] = acc[v];
    }
  }
}

// ------------------------------------------------------------------- zeros
__global__ void zero_kernel(float* __restrict__ p, int n) {
  int t = blockIdx.x * blockDim.x + threadIdx.x;
  if (t < n) p[t] = 0.f;
}

// -------------------------------------------- per-channel norm sum / sumsq
// Block per (o,b): scan S positions of p (layout b*bStride + o*3S + c*S + s).
__global__ void norm_stats_kernel(const float* __restrict__ P, float* __restrict__ stats,
                                  int Co, long S, long bStride) {
  const int o = blockIdx.x, b = blockIdx.y;
  const float* pb = P + (long)b * bStride + (long)o * 3 * S;
  float s = 0.f, sq = 0.f;
  for (long i = threadIdx.x; i < S; i += blockDim.x) {
    float px = pb[i], py = pb[S + i], pz = pb[2 * S + i];
    float nrm = sqrtf(px * px + py * py + pz * pz) + EPSV;
    s += nrm; sq += nrm * nrm;
  }
  __shared__ float sh[512];
  sh[threadIdx.x] = s; sh[threadIdx.x + 256] = sq;
  __syncthreads();
  for (int st = 128; st > 0; st >>= 1) {
    if ((int)threadIdx.x < st) {
      sh[threadIdx.x] += sh[threadIdx.x + st];
      sh[threadIdx.x + 256] += sh[threadIdx.x + 256 + st];
    }
    __syncthreads();
  }
  if (threadIdx.x == 0) {
    atomicAdd(&stats[o * 2 + 0], sh[0]);
    atomicAdd(&stats[o * 2 + 1], sh[256]);
  }
}

__global__ void bn_finalize_kernel(const float* __restrict__ stats, float* __restrict__ mv,
                                   int Co, float cnt) {
  int o = blockIdx.x * blockDim.x + threadIdx.x;
  if (o >= Co) return;
  float mean = stats[o * 2] / cnt;
  float var = stats[o * 2 + 1] / cnt - mean * mean;
  mv[o * 2] = mean;
  mv[o * 2 + 1] = 1.f / sqrtf(var + BNEPS);
}

// ------------------------------------------- VN batchnorm + directional act
__device__ __forceinline__ void vn_act_one(float px, float py, float pz,
                                           float dx, float dy, float dz,
                                           float g, float bt, float mean, float rstd,
                                           float& ox, float& oy, float& oz) {
  float nrm = sqrtf(px * px + py * py + pz * pz) + EPSV;
  float f = (g * (nrm - mean) * rstd + bt) / nrm;
  px *= f; py *= f; pz *= f;
  float dot = px * dx + py * dy + pz * dz;
  float qx = px, qy = py, qz = pz;
  if (dot < 0.f) {
    float s = dot / (dx * dx + dy * dy + dz * dz + EPSV);
    qx -= s * dx; qy -= s * dy; qz -= s * dz;
  }
  ox = NEGS * px + (1.f - NEGS) * qx;
  oy = NEGS * py + (1.f - NEGS) * qy;
  oz = NEGS * pz + (1.f - NEGS) * qz;
}

__global__ void vn_act_kernel(const float* __restrict__ P, const float* __restrict__ Dm,
                              long dStrideO, long bStride,
                              const float* __restrict__ mv,
                              const float* __restrict__ gamma, const float* __restrict__ beta,
                              float* __restrict__ out, long outBS,
                              int Co, long S, long total) {
  long t = (long)blockIdx.x * blockDim.x + threadIdx.x;
  if (t >= total) return;
  long s = t % S; long r = t / S;
  int o = (int)(r % Co), b = (int)(r / Co);
  const float* pp = P + (long)b * bStride + (long)o * 3 * S + s;
  const float* dd = Dm + (long)b * bStride + (long)o * dStrideO + s;
  float ox, oy, oz;
  vn_act_one(pp[0], pp[S], pp[2 * S], dd[0], dd[S], dd[2 * S],
             gamma[o], beta[o], mv[o * 2], mv[o * 2 + 1], ox, oy, oz);
  float* op = out + (long)b * outBS + (long)o * 3 * S + s;
  op[0] = ox; op[S] = oy; op[2 * S] = oz;
}

// Same, fused with mean over the Kn neighbor axis (S = N*Kn -> N).
__global__ void vn_act_meank_kernel(const float* __restrict__ P, const float* __restrict__ Dm,
                                    long dStrideO, long bStride,
                                    const float* __restrict__ mv,
                                    const float* __restrict__ gamma, const float* __restrict__ beta,
                                    float* __restrict__ out, long outBS, long outOS,
                                    int Co, int N, int Kn, long total) {
  long t = (long)blockIdx.x * blockDim.x + threadIdx.x;
  if (t >= total) return;
  int n = (int)(t % N); long r = t / N;
  int o = (int)(r % Co), b = (int)(r / Co);
  const long S = (long)N * Kn;
  const float* pp = P + (long)b * bStride + (long)o * 3 * S + (long)n * Kn;
  const float* dd = Dm + (long)b * bStride + (long)o * dStrideO + (long)n * Kn;
  float g = gamma[o], bt = beta[o], mean = mv[o * 2], rstd = mv[o * 2 + 1];
  float ax = 0.f, ay = 0.f, az = 0.f;
  for (int kk = 0; kk < Kn; ++kk) {
    float ox, oy, oz;
    vn_act_one(pp[kk], pp[S + kk], pp[2 * S + kk],
               dd[kk], dd[S + kk], dd[2 * S + kk], g, bt, mean, rstd, ox, oy, oz);
    ax += ox; ay += oy; az += oz;
  }
  float inv = 1.f / (float)Kn;
  float* op = out + (long)b * outBS + (long)o * outOS + n;
  op[0] = ax * inv; op[N] = ay * inv; op[2 * N] = az * inv;
}

// ------------------------- mean over N of h6 -> xg (duplicated halves) -----
__global__ void meanN_to_xg_kernel(const float* __restrict__ h, float* __restrict__ xg,
                                   int F2, int N, long hBS, int total) {
  int t = blockIdx.x * blockDim.x + threadIdx.x;
  if (t >= total) return;
  int c = t % 3; int r = t / 3;
  int o = r % F2; int b = r / F2;
  const float* p = h + (long)b * hBS + ((long)o * 3 + c) * N;
  float s = 0.f;
  for (int n = 0; n < N; ++n) s += p[n];
  s /= (float)N;
  xg[((long)b * (2 * F2) + o) * 3 + c] = s;
  xg[((long)b * (2 * F2) + F2 + o) * 3 + c] = s;
}

// ------------------------------------------------ x_std = xg (B,1024,3) . z0
__global__ void xstd_kernel(const float* __restrict__ xg, const float* __restrict__ z0,
                            float* __restrict__ xs, int total) {
  int t = blockIdx.x * blockDim.x + threadIdx.x;
  if (t >= total) return;
  int k = t % 3; int r = t / 3;
  int i = r % 1024; int b = r / 1024;
  const float* xr = xg + ((long)b * 1024 + i) * 3;
  const float* z = z0 + (long)b * 9;
  xs[t] = xr[0] * z[k] + xr[1] * z[3 + k] + xr[2] * z[6 + k];
}

// --------------------------- x1_out[b,i,o] = x_std[b,i,:] . W0[o,:] + b0[o]
__global__ void final_out_kernel(const float* __restrict__ xs, const float* __restrict__ W0,
                                 const float* __restrict__ b0, float* __restrict__ out,
                                 long total) {
  long t = (long)blockIdx.x * blockDim.x + threadIdx.x;
  if (t >= total) return;
  int o = (int)(t % 1024); long r = t / 1024;
  int i = (int)(r % 1024); int b = (int)(r / 1024);
  const float* x = xs + ((long)b * 1024 + i) * 3;
  const float* w = W0 + (long)o * 3;
  out[t] = x[0] * w[0] + x[1] * w[1] + x[2] * w[2] + b0[o];
}

// ===========================================================================
static inline long cdivl(long a, long b) { return (a + b - 1) / b; }

template <int KD>
static void launch_gemm(const float* A, const float* Bm, float* Cm, int M,
                        long Ncols, long ldb, long bBS, long cBS, int batches,
                        hipStream_t stream) {
  dim3 g((unsigned)cdivl(Ncols, 128), (unsigned)cdivl(M, 16), batches);
  gemm_wmma_kernel<KD><<<g, 256, 0, stream>>>(A, Bm, Cm, M, Ncols, ldb, bBS, cBS);
}

extern "C" void kernel_launch(void* const* d_in, const int* in_sizes, int n_in,
                              void* d_out, int out_size, void* d_ws, size_t ws_size,
                              hipStream_t stream) {
  (void)in_sizes; (void)n_in; (void)out_size; (void)ws_size;
  const int BATCH = 16, N = 2048, KN = 20;
  const long S_e = (long)N * KN;     // 40960 edge positions per point row
  const long C3S = 3 * S_e;          // 122880
  const long C3N = 3L * N;           // 6144
  const long XBS = 63L * C3N;        // x123 batch stride

  const float* X   = (const float*)d_in[0];
  const float *W1 = (const float*)d_in[1],  *Dd1 = (const float*)d_in[2],
              *G1 = (const float*)d_in[3],  *Bt1 = (const float*)d_in[4];
  const float *W2 = (const float*)d_in[5],  *Dd2 = (const float*)d_in[6],
              *G2 = (const float*)d_in[7],  *Bt2 = (const float*)d_in[8];
  const float *W3 = (const float*)d_in[9],  *Dd3 = (const float*)d_in[10],
              *G3 = (const float*)d_in[11], *Bt3 = (const float*)d_in[12];
  const float *W4 = (const float*)d_in[13], *Dd4 = (const float*)d_in[14],
              *G4 = (const float*)d_in[15], *Bt4 = (const float*)d_in[16];
  const float *W5 = (const float*)d_in[17], *Dd5 = (const float*)d_in[18],
              *G5 = (const float*)d_in[19], *Bt5 = (const float*)d_in[20];
  const float *W6 = (const float*)d_in[21], *Dd6 = (const float*)d_in[22],
              *G6 = (const float*)d_in[23], *Bt6 = (const float*)d_in[24];
  const float *Wv1 = (const float*)d_in[25], *Dv1 = (const float*)d_in[26],
              *Gv1 = (const float*)d_in[27], *Bv1 = (const float*)d_in[28];
  const float *Wv2 = (const float*)d_in[29], *Dv2 = (const float*)d_in[30],
              *Gv2 = (const float*)d_in[31], *Bv2 = (const float*)d_in[32];
  const float *Wz = (const float*)d_in[33], *W0 = (const float*)d_in[34],
              *B0 = (const float*)d_in[35];

  // ------------------------------ workspace layout (~0.9 GB, 256B aligned)
  size_t off = 0;
  auto alloc = [&](size_t bytes) -> char* {
    char* p = (char*)d_ws + off;
    off += (bytes + 255) & ~(size_t)255;
    return p;
  };
  int*   g_idx  = (int*)  alloc((size_t)BATCH * N * KN * 4);
  float* g_xx   = (float*)alloc((size_t)BATCH * N * 4);
  float* g_A    = (float*)alloc((size_t)1024 * 1024 * 4);
  float* g_st   = (float*)alloc((size_t)2 * 1024 * 4);
  float* g_mv   = (float*)alloc((size_t)2 * 1024 * 4);
  float* g_H0   = (float*)alloc((size_t)BATCH * 42 * C3S * 4);   // gather buf
  float* g_PD   = (float*)alloc((size_t)BATCH * 42 * C3S * 4);   // packed P|D
  float* g_H    = (float*)alloc((size_t)BATCH * 512 * C3N * 4);  // act out / h6
  float* g_X123 = (float*)alloc((size_t)BATCH * 63 * C3N * 4);   // x1|x2|x3
  float* g_PDz  = (float*)alloc((size_t)BATCH * 1024 * 3 * 4);
  float* g_z1   = (float*)alloc((size_t)BATCH * 512 * 3 * 4);
  float* g_PDz2 = (float*)alloc((size_t)BATCH * 512 * 3 * 4);
  float* g_z2   = (float*)alloc((size_t)BATCH * 256 * 3 * 4);
  float* g_z0   = (float*)alloc((size_t)BATCH * 9 * 4);
  float* g_xstd = (float*)alloc((size_t)BATCH * 1024 * 3 * 4);

  auto bnstats = [&](const float* P, int Co, long S, long bStride, float cnt) {
    zero_kernel<<<(unsigned)cdivl(2 * Co, 256), 256, 0, stream>>>(g_st, 2 * Co);
    norm_stats_kernel<<<dim3(Co, BATCH), 256, 0, stream>>>(P, g_st, Co, S, bStride);
    bn_finalize_kernel<<<(unsigned)cdivl(Co, 256), 256, 0, stream>>>(g_st, g_mv, Co, cnt);
  };

  // ============================ Stage 1 ====================================
  {
    long tot = (long)BATCH * N;
    row_sumsq_kernel<<<(unsigned)cdivl(tot, 256), 256, 0, stream>>>(X, g_xx, 3, N, 3L * N, BATCH);
    knn_topk_kernel<3><<<dim3(N / 16, BATCH), 128, (size_t)16 * N * 4, stream>>>(
        X, g_xx, g_idx, N, 3L * N);
  }
  {
    long tot = (long)BATCH * 2 * 3 * S_e;
    graph_feature_kernel<<<(unsigned)cdivl(tot, 256), 256, 0, stream>>>(
        X, g_idx, g_H0, 1, N, KN, 3L * N, tot);
  }
  // block 1: Co=21, K=2
  packA_kernel<<<1, 128, 0, stream>>>(W1, Dd1, g_A, 21, 21, 2);
  launch_gemm<2>(g_A, g_H0, g_PD, 42, C3S, C3S, 2 * C3S, 42 * C3S, BATCH, stream);
  bnstats(g_PD, 21, S_e, 42 * C3S, (float)((long)BATCH * S_e));
  {
    long tot = (long)BATCH * 21 * S_e;
    vn_act_kernel<<<(unsigned)cdivl(tot, 256), 256, 0, stream>>>(
        g_PD, g_PD + 21 * C3S, C3S, 42 * C3S, g_mv, G1, Bt1, g_H, 21 * C3S, 21, S_e, tot);
  }
  // block 2: Co=21, K=21, fuse mean over k -> x1 (rows 0..20 of x123)
  packA_kernel<<<(unsigned)cdivl(42 * 21, 256), 256, 0, stream>>>(W2, Dd2, g_A, 21, 21, 21);
  launch_gemm<21>(g_A, g_H, g_PD, 42, C3S, C3S, 21 * C3S, 42 * C3S, BATCH, stream);
  bnstats(g_PD, 21, S_e, 42 * C3S, (float)((long)BATCH * S_e));
  {
    long tot = (long)BATCH * 21 * N;
    vn_act_meank_kernel<<<(unsigned)cdivl(tot, 256), 256, 0, stream>>>(
        g_PD, g_PD + 21 * C3S, C3S, 42 * C3S, g_mv, G2, Bt2,
        g_X123, XBS, C3N, 21, N, KN, tot);
  }

  // ============================ Stage 2 ====================================
  {
    long tot = (long)BATCH * N;
    row_sumsq_kernel<<<(unsigned)cdivl(tot, 256), 256, 0, stream>>>(g_X123, g_xx, 63, N, XBS, BATCH);
    knn_topk_kernel<63><<<dim3(N / 16, BATCH), 128, (size_t)16 * N * 4, stream>>>(
        g_X123, g_xx, g_idx, N, XBS);
  }
  {
    long tot = (long)BATCH * 42 * 3 * S_e;
    graph_feature_kernel<<<(unsigned)cdivl(tot, 256), 256, 0, stream>>>(
        g_X123, g_idx, g_H0, 21, N, KN, XBS, tot);
  }
  packA_kernel<<<(unsigned)cdivl(42 * 42, 256), 256, 0, stream>>>(W3, Dd3, g_A, 21, 21, 42);
  launch_gemm<42>(g_A, g_H0, g_PD, 42, C3S, C3S, 42 * C3S, 42 * C3S, BATCH, stream);
  bnstats(g_PD, 21, S_e, 42 * C3S, (float)((long)BATCH * S_e));
  {
    long tot = (long)BATCH * 21 * S_e;
    vn_act_kernel<<<(unsigned)cdivl(tot, 256), 256, 0, stream>>>(
        g_PD, g_PD + 21 * C3S, C3S, 42 * C3S, g_mv, G3, Bt3, g_H, 21 * C3S, 21, S_e, tot);
  }
  packA_kernel<<<(unsigned)cdivl(42 * 21, 256), 256, 0, stream>>>(W4, Dd4, g_A, 21, 21, 21);
  launch_gemm<21>(g_A, g_H, g_PD, 42, C3S, C3S, 21 * C3S, 42 * C3S, BATCH, stream);
  bnstats(g_PD, 21, S_e, 42 * C3S, (float)((long)BATCH * S_e));
  {
    long tot = (long)BATCH * 21 * N;   // -> x2 (rows 21..41)
    vn_act_meank_kernel<<<(unsigned)cdivl(tot, 256), 256, 0, stream>>>(
        g_PD, g_PD + 21 * C3S, C3S, 42 * C3S, g_mv, G4, Bt4,
        g_X123 + 21 * C3N, XBS, C3N, 21, N, KN, tot);
  }

  // ============================ Stage 3 ====================================
  {
    long tot = (long)BATCH * N;
    row_sumsq_kernel<<<(unsigned)cdivl(tot, 256), 256, 0, stream>>>(
        g_X123 + 21 * C3N, g_xx, 63, N, XBS, BATCH);
    knn_topk_kernel<63><<<dim3(N / 16, BATCH), 128, (size_t)16 * N * 4, stream>>>(
        g_X123 + 21 * C3N, g_xx, g_idx, N, XBS);
  }
  {
    long tot = (long)BATCH * 42 * 3 * S_e;
    graph_feature_kernel<<<(unsigned)cdivl(tot, 256), 256, 0, stream>>>(
        g_X123 + 21 * C3N, g_idx, g_H0, 21, N, KN, XBS, tot);
  }
  packA_kernel<<<(unsigned)cdivl(42 * 42, 256), 256, 0, stream>>>(W5, Dd5, g_A, 21, 21, 42);
  launch_gemm<42>(g_A, g_H0, g_PD, 42, C3S, C3S, 42 * C3S, 42 * C3S, BATCH, stream);
  bnstats(g_PD, 21, S_e, 42 * C3S, (float)((long)BATCH * S_e));
  {
    long tot = (long)BATCH * 21 * N;   // -> x3 (rows 42..62)
    vn_act_meank_kernel<<<(unsigned)cdivl(tot, 256), 256, 0, stream>>>(
        g_PD, g_PD + 21 * C3S, C3S, 42 * C3S, g_mv, G5, Bt5,
        g_X123 + 42 * C3N, XBS, C3N, 21, N, KN, tot);
  }

  // ============================ Block 6 (point domain) =====================
  packA_kernel<<<(unsigned)cdivl(513 * 63, 256), 256, 0, stream>>>(W6, Dd6, g_A, 512, 1, 63);
  launch_gemm<63>(g_A, g_X123, g_PD, 513, C3N, C3N, XBS, 513 * C3N, BATCH, stream);
  bnstats(g_PD, 512, (long)N, 513 * C3N, (float)((long)BATCH * N));
  {
    long tot = (long)BATCH * 512 * N;   // d (1 row) broadcast over channels
    vn_act_kernel<<<(unsigned)cdivl(tot, 256), 256, 0, stream>>>(
        g_PD, g_PD + 512 * C3N, 0, 513 * C3N, g_mv, G6, Bt6, g_H, 512 * C3N, 512, (long)N, tot);
  }
  // xg = [mean_N h ; mean_N h] -> first output region of d_out
  float* xg = (float*)d_out;
  {
    int tot = BATCH * 512 * 3;
    meanN_to_xg_kernel<<<(unsigned)cdivl(tot, 256), 256, 0, stream>>>(
        g_H, xg, 512, N, 512 * C3N, tot);
  }

  // ============================ z path =====================================
  packA_kernel<<<(unsigned)cdivl(1024 * 1024, 256), 256, 0, stream>>>(Wv1, Dv1, g_A, 512, 512, 1024);
  launch_gemm<1024>(g_A, xg, g_PDz, 1024, 3, 3, 1024 * 3, 1024 * 3, BATCH, stream);
  bnstats(g_PDz, 512, 1, 1024 * 3, (float)BATCH);
  {
    long tot = (long)BATCH * 512;
    vn_act_kernel<<<(unsigned)cdivl(tot, 256), 256, 0, stream>>>(
        g_PDz, g_PDz + 512 * 3, 3, 1024 * 3, g_mv, Gv1, Bv1, g_z1, 512 * 3, 512, 1, tot);
  }
  packA_kernel<<<(unsigned)cdivl(512 * 512, 256), 256, 0, stream>>>(Wv2, Dv2, g_A, 256, 256, 512);
  launch_gemm<512>(g_A, g_z1, g_PDz2, 512, 3, 3, 512 * 3, 512 * 3, BATCH, stream);
  bnstats(g_PDz2, 256, 1, 512 * 3, (float)BATCH);
  {
    long tot = (long)BATCH * 256;
    vn_act_kernel<<<(unsigned)cdivl(tot, 256), 256, 0, stream>>>(
        g_PDz2, g_PDz2 + 256 * 3, 3, 512 * 3, g_mv, Gv2, Bv2, g_z2, 256 * 3, 256, 1, tot);
  }
  // z0 = Wz (3x256) . z2 (B,256,3) -> (B,3,3)
  launch_gemm<256>(Wz, g_z2, g_z0, 3, 3, 3, 256 * 3, 9, BATCH, stream);
  {
    int tot = BATCH * 1024 * 3;
    xstd_kernel<<<(unsigned)cdivl(tot, 256), 256, 0, stream>>>(xg, g_z0, g_xstd, tot);
  }
  {
    long tot = (long)BATCH * 1024 * 1024;
    final_out_kernel<<<(unsigned)cdivl(tot, 256), 256, 0, stream>>>(
        g_xstd, W0, B0, (float*)d_out + (long)BATCH * 1024 * 3, tot);
  }
}